// GAT_88639535055056
// MI455X (gfx1250) — compile-verified
//
#include <hip/hip_runtime.h>
#include <hip/hip_bf16.h>

// ---------------- problem constants (match reference) ----------------
#define N_NODES   50000
#define N_EDGES   800000
#define N_FEAT    512
#define N_HEADS   8
#define OUT_PH    32
#define C_OUT     (N_HEADS * OUT_PH)   // 256
#define ALPHA     0.2f
#define ROWTILES  (N_NODES / 16)       // 3125, exact

typedef __attribute__((ext_vector_type(2))) float v2f;
typedef __attribute__((ext_vector_type(8))) float v8f;

// ---------------------------------------------------------------------
// Kernel 0: init out = 0, m = -inf, denom = 0
// ---------------------------------------------------------------------
__global__ __launch_bounds__(256) void gat_init(float* __restrict__ out,
                                                float* __restrict__ m,
                                                float* __restrict__ denom) {
    int t = blockIdx.x * blockDim.x + threadIdx.x;
    if (t < N_NODES * C_OUT) out[t] = 0.0f;
    if (t < N_NODES * N_HEADS) {
        m[t] = -__builtin_huge_valf();
        denom[t] = 0.0f;
    }
}

// ---------------------------------------------------------------------
// Kernel 1: Wh = h @ W' + Wb  via V_WMMA_F32_16X16X4_F32
//   B panel (16 cols x 512 K = 32 KB) staged in LDS once per block,
//   re-used by 8 waves each computing a different 16x16 row tile.
//   LDS layout pair-interleaved so each lane's (K,K+1) fragment is a
//   contiguous 8-byte ds_load.
// grid = (ceil(ROWTILES/8)=391, 16 colTiles), block = 256 (8 wave32)
// ---------------------------------------------------------------------
__global__ __launch_bounds__(256) void gat_wmma_gemm(const float* __restrict__ h,
                                                     const float* __restrict__ W,
                                                     const float* __restrict__ Wb,
                                                     float* __restrict__ Wh) {
    __shared__ __align__(16) float Bs[N_FEAT * 16];   // 32 KB

    const int tid     = threadIdx.x;
    const int colTile = blockIdx.y;          // 0..15
    const int head    = colTile >> 1;        // 16 | 32 -> col tile inside one head
    const int obase   = (colTile & 1) * 16;

    // ---- stage B[f][colTile*16 + col] into LDS, pair-interleaved ----
    {
        const int col = tid & 15;
        const int f0  = tid >> 4;            // 0..15
        const float* Wsrc = W + (size_t)head * N_FEAT * OUT_PH + obase + col;
        #pragma unroll
        for (int i = 0; i < N_FEAT / 16; ++i) {      // 32 iters
            const int f = f0 + i * 16;
            // B[k][col] lives at Bs[(k>>1)*32 + col*2 + (k&1)]
            Bs[((f >> 1) * 16 + col) * 2 + (f & 1)] = Wsrc[(size_t)f * OUT_PH];
        }
    }
    __syncthreads();

    const int lane    = tid & 31;
    const int wave    = tid >> 5;
    const int rowTile = blockIdx.x * 8 + wave;       // wave-uniform guard
    if (rowTile >= ROWTILES) return;

    const int half = lane >> 4;      // 0: K pair {0,1}, 1: K pair {2,3}
    const int mrow = lane & 15;      // A row / B-D column within tile
    const int rowBase = rowTile * 16;
    const int c = colTile * 16 + mrow;               // lane's output column

    const float* Arow = h + (size_t)(rowBase + mrow) * N_FEAT;
    const v2f*   Bp   = (const v2f*)Bs;              // Bp[(k>>1)*16 + col]

    v8f acc = {0.f, 0.f, 0.f, 0.f, 0.f, 0.f, 0.f, 0.f};

    #pragma unroll 8
    for (int k0 = 0; k0 < N_FEAT; k0 += 4) {
        const int ka = k0 + 2 * half;                // even
        const v2f afrag = *(const v2f*)(Arow + ka);  // global_load_b64
        const v2f bfrag = Bp[(ka >> 1) * 16 + mrow]; // ds_load_b64
        acc = __builtin_amdgcn_wmma_f32_16x16x4_f32(
            /*neg_a=*/false, afrag, /*neg_b=*/false, bfrag,
            /*c_mod=*/(short)0, acc, /*reuse_a=*/false, /*reuse_b=*/false);
    }

    const float bias = Wb[c];        // column constant across the lane's 8 rows
    #pragma unroll
    for (int j = 0; j < 8; ++j) {
        const int row = rowBase + j + (half << 3);   // M = j (lanes 0-15), 8+j (16-31)
        Wh[(size_t)row * C_OUT + c] = acc[j] + bias;
    }
}

// ---------------------------------------------------------------------
// Kernel 2: per-node attention logits
// ---------------------------------------------------------------------
__global__ __launch_bounds__(256) void gat_scores(const float* __restrict__ Wh,
                                                  const float* __restrict__ a,
                                                  float* __restrict__ e_s,
                                                  float* __restrict__ e_d) {
    int t = blockIdx.x * blockDim.x + threadIdx.x;
    if (t >= N_NODES * N_HEADS) return;
    const int n  = t >> 3;
    const int hh = t & 7;
    const float* row = Wh + (size_t)n * C_OUT + hh * OUT_PH;
    const float* as  = a + hh * 2 * OUT_PH;
    const float* ad  = as + OUT_PH;
    float s = 0.f, d = 0.f;
    #pragma unroll
    for (int k = 0; k < OUT_PH; ++k) {
        const float v = row[k];
        s += v * as[k];
        d += v * ad[k];
    }
    e_s[t] = s;
    e_d[t] = d;
}

// ---------------------------------------------------------------------
// float atomic max via CAS (GLOBAL_ATOMIC_CMPSWAP_B32)
// ---------------------------------------------------------------------
__device__ inline void atomicMaxFloat(float* addr, float val) {
    int* ai = (int*)addr;
    int old = __float_as_int(*addr);
    while (__int_as_float(old) < val) {
        const int assumed = old;
        old = atomicCAS(ai, assumed, __float_as_int(val));
        if (old == assumed) break;
    }
}

// ---------------------------------------------------------------------
// Kernel 3: e = leakyrelu(e_s[src]+e_d[dst]+ab); store e; segment max
// ---------------------------------------------------------------------
__global__ __launch_bounds__(256) void gat_edge_max(const int* __restrict__ src,
                                                    const int* __restrict__ dst,
                                                    const float* __restrict__ e_s,
                                                    const float* __restrict__ e_d,
                                                    const float* __restrict__ ab,
                                                    float* __restrict__ ebuf,
                                                    float* __restrict__ m) {
    int t = blockIdx.x * blockDim.x + threadIdx.x;
    if (t >= N_EDGES * N_HEADS) return;
    const int e  = t >> 3;
    const int hh = t & 7;
    const int si = src[e], di = dst[e];
    float x = e_s[si * N_HEADS + hh] + e_d[di * N_HEADS + hh] + ab[hh];
    x = x > 0.f ? x : ALPHA * x;     // LeakyReLU
    ebuf[t] = x;
    atomicMaxFloat(&m[di * N_HEADS + hh], x);
}

// ---------------------------------------------------------------------
// Kernel 4: w = exp(e - m[dst]); store w in place; denom[dst] += w
// ---------------------------------------------------------------------
__global__ __launch_bounds__(256) void gat_edge_denom(const int* __restrict__ dst,
                                                      const float* __restrict__ m,
                                                      float* __restrict__ wbuf,
                                                      float* __restrict__ denom) {
    int t = blockIdx.x * blockDim.x + threadIdx.x;
    if (t >= N_EDGES * N_HEADS) return;
    const int e  = t >> 3;
    const int hh = t & 7;
    const int di = dst[e];
    const float w = __expf(wbuf[t] - m[di * N_HEADS + hh]);
    wbuf[t] = w;
    atomicAdd(&denom[di * N_HEADS + hh], w);
}

// ---------------------------------------------------------------------
// Kernel 5: out[dst, c] += (w/denom') * Wh[src, c]
// one thread per (edge, c); lanes cover consecutive c -> coalesced
// gather of Wh[src,*] and coalesced f32 atomics into out[dst,*].
// ---------------------------------------------------------------------
__global__ __launch_bounds__(256) void gat_edge_scatter(const int* __restrict__ src,
                                                        const int* __restrict__ dst,
                                                        const float* __restrict__ wbuf,
                                                        const float* __restrict__ denom,
                                                        const float* __restrict__ Wh,
                                                        float* __restrict__ out) {
    int t = blockIdx.x * blockDim.x + threadIdx.x;   // < 204,800,000 (fits int)
    if (t >= N_EDGES * C_OUT) return;
    const int e  = t >> 8;        // / C_OUT
    const int c  = t & 255;
    const int hh = c >> 5;
    const int si = src[e], di = dst[e];
    const float w  = wbuf[e * N_HEADS + hh];
    const float dn = denom[di * N_HEADS + hh];
    const float att = w / (dn > 0.f ? dn : 1.f);
    atomicAdd(&out[(size_t)di * C_OUT + c], att * Wh[(size_t)si * C_OUT + c]);
}

// ---------------------------------------------------------------------
extern "C" void kernel_launch(void* const* d_in, const int* in_sizes, int n_in,
                              void* d_out, int out_size, void* d_ws, size_t ws_size,
                              hipStream_t stream) {
    const float* h   = (const float*)d_in[0];   // [N_NODES, N_FEAT]
    const int*   src = (const int*)  d_in[1];   // [N_EDGES]
    const int*   dst = (const int*)  d_in[2];   // [N_EDGES]
    const float* W   = (const float*)d_in[3];   // [H, F, O]
    const float* Wb  = (const float*)d_in[4];   // [H, O]
    const float* a   = (const float*)d_in[5];   // [H, 2*O]
    const float* ab  = (const float*)d_in[6];   // [H]
    float* out = (float*)d_out;                 // [N_NODES, C_OUT]

    // workspace partition (floats): 12.8M + 0.4M*4 + 6.4M = 20.8M = 83.2 MB
    float* ws    = (float*)d_ws;
    float* Wh    = ws;                                   // 12,800,000
    float* e_s   = Wh   + (size_t)N_NODES * C_OUT;       //    400,000
    float* e_d   = e_s  + (size_t)N_NODES * N_HEADS;     //    400,000
    float* m     = e_d  + (size_t)N_NODES * N_HEADS;     //    400,000
    float* denom = m    + (size_t)N_NODES * N_HEADS;     //    400,000
    float* wbuf  = denom + (size_t)N_NODES * N_HEADS;    //  6,400,000 (e, then w)

    // 0) init accumulators
    {
        const int n = N_NODES * C_OUT;
        gat_init<<<(n + 255) / 256, 256, 0, stream>>>(out, m, denom);
    }
    // 1) Wh = h @ W + Wb  (WMMA f32 16x16x4, LDS-staged B)
    {
        dim3 grid((ROWTILES + 7) / 8, 16, 1);
        gat_wmma_gemm<<<grid, 256, 0, stream>>>(h, W, Wb, Wh);
    }
    // 2) per-node logits
    {
        const int n = N_NODES * N_HEADS;
        gat_scores<<<(n + 255) / 256, 256, 0, stream>>>(Wh, a, e_s, e_d);
    }
    // 3) edge logits + segment max
    {
        const int n = N_EDGES * N_HEADS;
        gat_edge_max<<<(n + 255) / 256, 256, 0, stream>>>(src, dst, e_s, e_d, ab, wbuf, m);
    }
    // 4) w = exp(e - m), segment sum
    {
        const int n = N_EDGES * N_HEADS;
        gat_edge_denom<<<(n + 255) / 256, 256, 0, stream>>>(dst, m, wbuf, denom);
    }
    // 5) weighted scatter into out
    {
        const int n = N_EDGES * C_OUT;
        gat_edge_scatter<<<(n + 255) / 256, 256, 0, stream>>>(src, dst, wbuf, denom, Wh, out);
    }
}